// Grouper5_2903397892783
// MI455X (gfx1250) — compile-verified
//
#include <hip/hip_runtime.h>

typedef float v2f __attribute__((ext_vector_type(2)));
typedef float v8f __attribute__((ext_vector_type(8)));

#define BB 4
#define NN 8192
#define MM 2048
#define KK 32
#define CC 64
#define RAD2 (0.4f * 0.4f)
#define WPB 2   // waves (queries) per block

// featT: [B][N][C] transposed features for coalesced neighbor gather
__global__ void transpose_feat_kernel(const float* __restrict__ f, float* __restrict__ ft) {
    size_t t = (size_t)blockIdx.x * blockDim.x + threadIdx.x;
    if (t >= (size_t)BB * CC * NN) return;
    int n = (int)(t % NN);
    int c = (int)((t / NN) % CC);
    int b = (int)(t / ((size_t)NN * CC));
    ft[((size_t)b * NN + n) * CC + c] = f[t];
}

__global__ __launch_bounds__(WPB * 32) void grouper_fused_kernel(
    const float* __restrict__ xyz, const float* __restrict__ new_xyz,
    const float* __restrict__ features, const float* __restrict__ qfeat,
    const float* __restrict__ qW, const float* __restrict__ qb,
    const float* __restrict__ afW, const float* __restrict__ afb,
    const float* __restrict__ axW, const float* __restrict__ axb,
    const float* __restrict__ mW1, const float* __restrict__ mb1,
    const float* __restrict__ mW2, const float* __restrict__ mb2,
    const float* __restrict__ xW, const float* __restrict__ xbias,
    const float* __restrict__ featT, float* __restrict__ out)
{
    __shared__ float Xt[WPB][32 * 66];   // g_feat, [col][c] stride 66
    __shared__ float Hb[WPB][32 * 66];   // hidden relu(mW1@X+mb1), same layout
    __shared__ float vbuf[WPB][CC];      // afW@q + afb
    __shared__ float qfb[WPB][CC];
    __shared__ float qtb[WPB][CC];
    __shared__ int   idxb[WPB][KK];
    __shared__ float mb1b[CC], mb2b[CC], axbb[CC], axw0b[CC], axw1b[CC], axw2b[CC];

    const int tid  = threadIdx.x;
    const int wi   = tid >> 5;
    const int lane = tid & 31;
    const int li   = lane & 15;
    const int g    = lane >> 4;
    const int wq   = blockIdx.x * WPB + wi;   // global query id
    const int b    = wq / MM;
    const int m    = wq % MM;

    if (tid < CC) {
        mb1b[tid]  = mb1[tid];
        mb2b[tid]  = mb2[tid];
        axbb[tid]  = axb[tid];
        axw0b[tid] = axW[tid * 3 + 0];
        axw1b[tid] = axW[tid * 3 + 1];
        axw2b[tid] = axW[tid * 3 + 2];
    }

    // ---------------- ball query: first KK ascending in-ball indices -------
    const float qx = new_xyz[(b * MM + m) * 3 + 0];
    const float qy = new_xyz[(b * MM + m) * 3 + 1];
    const float qz = new_xyz[(b * MM + m) * 3 + 2];
    const float* xyzb = xyz + (size_t)b * NN * 3;

    int cnt = 0;
    for (int n0 = 0; n0 < NN; n0 += 32) {
        int n = n0 + lane;
        float px = xyzb[n * 3 + 0];
        float py = xyzb[n * 3 + 1];
        float pz = xyzb[n * 3 + 2];
        float dx = px - qx, dy = py - qy, dz = pz - qz;
        bool within = (dx * dx + dy * dy + dz * dz) < RAD2;
        unsigned mask = (unsigned)__ballot(within);
        int slot = cnt + __popc(mask & ((1u << lane) - 1u));
        if (within && slot < KK) idxb[wi][slot] = n;
        cnt += __popc(mask);
        if (cnt >= KK) break;   // wave-uniform early exit
    }
    __syncthreads();

    const int  cc = cnt < KK ? cnt : KK;
    const bool rowvalid = cnt > 0;
    int   myidx;
    float myidn;
    if (!rowvalid) { myidx = NN - 1; myidn = 0.f; }
    else {
        int first = idxb[wi][0];
        myidx = (lane < cc) ? idxb[wi][lane] : first;
        myidn = (lane < cc) ? 1.f : 0.f;
    }

    // per-lane (= per-neighbor) relative coords & IDW weight
    const float gx = xyzb[myidx * 3 + 0] - qx;
    const float gy = xyzb[myidx * 3 + 1] - qy;
    const float gz = xyzb[myidx * 3 + 2] - qz;

    float dist = sqrtf(gx * gx + gy * gy + gz * gz);
    float rcp  = 1.f / (dist + 1e-8f);
    float idn_div = rowvalid ? myidn : 1.f;
    rcp = rcp / idn_div;                         // literal ref semantics (inf for padded)
    float norm = rcp;
    norm += __shfl_xor(norm, 1, 32);
    norm += __shfl_xor(norm, 2, 32);
    norm += __shfl_xor(norm, 4, 32);
    norm += __shfl_xor(norm, 8, 32);
    norm += __shfl_xor(norm, 16, 32);
    const float w = (rcp / norm) * (rowvalid ? 1.f : 0.f);

    // ---------------- gather g_feat into LDS: Xt[col][c] -------------------
    float* X = Xt[wi];
    if (featT) {
        const float* src = featT + ((size_t)b * NN + myidx) * CC;
        #pragma unroll
        for (int c0 = 0; c0 < CC; c0 += 4) {
            float4 t = *(const float4*)(src + c0);
            v2f lo = {t.x, t.y};
            v2f hi = {t.z, t.w};
            *(v2f*)(&X[lane * 66 + c0])     = lo;
            *(v2f*)(&X[lane * 66 + c0 + 2]) = hi;
        }
    } else {
        const float* src = features + (size_t)b * CC * NN + myidx;
        for (int c = 0; c < CC; c++) X[lane * 66 + c] = src[(size_t)c * NN];
    }

    // stage query features
    qfb[wi][lane]      = qfeat[(size_t)b * CC * MM + (size_t)lane * MM + m];
    qfb[wi][lane + 32] = qfeat[(size_t)b * CC * MM + (size_t)(lane + 32) * MM + m];
    __syncthreads();

    // q = relu(qW @ qf + qb)   (each lane: rows lane, lane+32)
    #pragma unroll
    for (int h = 0; h < 2; h++) {
        int o = lane + 32 * h;
        float s = qb[o];
        for (int i = 0; i < CC; i++) s += qW[o * CC + i] * qfb[wi][i];
        qtb[wi][o] = fmaxf(s, 0.f);
    }
    __syncthreads();

    // vb = afW @ q + afb   (fold bias; feats_dist = relu(vb - afW@g_feat))
    #pragma unroll
    for (int h = 0; h < 2; h++) {
        int o = lane + 32 * h;
        float s = afb[o];
        for (int i = 0; i < CC; i++) s += afW[o * CC + i] * qtb[wi][i];
        vbuf[wi][o] = s;
    }
    __syncthreads();   // X, vbuf ready

    // ---------------- WMMA GEMMs (fp32 16x16x4) ----------------------------
    // A 16x4: lane (g,li) holds W[mt*16+li][ks*4+2g .. +1]
    // B 4x16: lane (g,li) holds Xop[ks*4+2g .. +1][nt*16+li]
    auto gemm_tile = [&](const float* __restrict__ W, const float* __restrict__ Bop,
                         int mt, int nt) -> v8f {
        v8f acc = {};
        const float* ab = W + (mt * 16 + li) * CC + 2 * g;
        const float* bb = Bop + (nt * 16 + li) * 66 + 2 * g;
        #pragma unroll
        for (int ks = 0; ks < 16; ks++) {
            v2f a  = *(const v2f*)(ab + ks * 4);
            v2f bf = *(const v2f*)(bb + ks * 4);
            acc = __builtin_amdgcn_wmma_f32_16x16x4_f32(
                false, a, false, bf, (short)0, acc, false, false);
        }
        return acc;
    };

    // GEMM1: H = relu(mW1 @ X + mb1) -> Hb
    for (int mt = 0; mt < 4; mt++)
        for (int nt = 0; nt < 2; nt++) {
            v8f h = gemm_tile(mW1, X, mt, nt);
            float* hp = &Hb[wi][(nt * 16 + li) * 66 + mt * 16 + 8 * g];
            #pragma unroll
            for (int r = 0; r < 8; r++)
                hp[r] = fmaxf(h[r] + mb1b[mt * 16 + 8 * g + r], 0.f);
        }
    __syncthreads();

    // GEMM2 (afW@X) + GEMM3 (mW2@H) + fused epilogue -> nf (channels 16..79)
    for (int mt = 0; mt < 4; mt++) {
        float nfacc[8] = {0.f, 0.f, 0.f, 0.f, 0.f, 0.f, 0.f, 0.f};
        for (int nt = 0; nt < 2; nt++) {
            v8f fda = gemm_tile(afW, X, mt, nt);
            v8f fa  = gemm_tile(mW2, Hb[wi], mt, nt);
            int src = nt * 16 + li;          // this lane's column
            float wc  = __shfl(w,  src, 32);
            float gxc = __shfl(gx, src, 32);
            float gyc = __shfl(gy, src, 32);
            float gzc = __shfl(gz, src, 32);
            #pragma unroll
            for (int r = 0; r < 8; r++) {
                int c = mt * 16 + 8 * g + r;
                float fd = fmaxf(vbuf[wi][c] - fda[r], 0.f);                    // feats_dist
                float F  = fmaxf(fa[r] + mb2b[c], 0.f);                         // nf pre-weight
                float sd = fmaxf(axw0b[c] * gxc + axw1b[c] * gyc +
                                 axw2b[c] * gzc + axbb[c], 0.f);                // space_dist
                nfacc[r] += wc * F * fd * sd;
            }
        }
        #pragma unroll
        for (int r = 0; r < 8; r++) {
            float s = nfacc[r];
            s += __shfl_xor(s, 1, 32);
            s += __shfl_xor(s, 2, 32);
            s += __shfl_xor(s, 4, 32);
            s += __shfl_xor(s, 8, 32);
            if (li == 0) {
                int c = mt * 16 + 8 * g + r;
                out[((size_t)b * 80 + 16 + c) * MM + m] = s;
            }
        }
    }

    // nx = max_k relu(xW @ g_xyz + xb) -> channels 0..15
    #pragma unroll
    for (int j = 0; j < 16; j++) {
        float t = fmaxf(xW[j * 3 + 0] * gx + xW[j * 3 + 1] * gy +
                        xW[j * 3 + 2] * gz + xbias[j], 0.f);
        t = fmaxf(t, __shfl_xor(t, 1, 32));
        t = fmaxf(t, __shfl_xor(t, 2, 32));
        t = fmaxf(t, __shfl_xor(t, 4, 32));
        t = fmaxf(t, __shfl_xor(t, 8, 32));
        t = fmaxf(t, __shfl_xor(t, 16, 32));
        if (lane == 0) out[((size_t)b * 80 + j) * MM + m] = t;
    }
}

extern "C" void kernel_launch(void* const* d_in, const int* in_sizes, int n_in,
                              void* d_out, int out_size, void* d_ws, size_t ws_size,
                              hipStream_t stream) {
    const float* xyz      = (const float*)d_in[0];
    const float* new_xyz  = (const float*)d_in[1];
    const float* features = (const float*)d_in[2];
    const float* qfeat    = (const float*)d_in[3];
    const float* qW  = (const float*)d_in[4];
    const float* qb  = (const float*)d_in[5];
    const float* afW = (const float*)d_in[6];
    const float* afb = (const float*)d_in[7];
    const float* axW = (const float*)d_in[8];
    const float* axb = (const float*)d_in[9];
    const float* mW1 = (const float*)d_in[10];
    const float* mb1 = (const float*)d_in[11];
    const float* mW2 = (const float*)d_in[12];
    const float* mb2 = (const float*)d_in[13];
    const float* xW  = (const float*)d_in[14];
    const float* xb  = (const float*)d_in[15];
    float* outp = (float*)d_out;

    // output[0]: new_xyz passthrough (B*M*3 floats)
    hipMemcpyAsync(outp, d_in[1], (size_t)BB * MM * 3 * sizeof(float),
                   hipMemcpyDeviceToDevice, stream);

    // optional feature transpose in workspace for coalesced gathers
    float* featT = nullptr;
    size_t need = (size_t)BB * NN * CC * sizeof(float);
    if (ws_size >= need) {
        featT = (float*)d_ws;
        int total = BB * CC * NN;
        transpose_feat_kernel<<<(total + 255) / 256, 256, 0, stream>>>(features, featT);
    }

    grouper_fused_kernel<<<BB * MM / WPB, WPB * 32, 0, stream>>>(
        xyz, new_xyz, features, qfeat, qW, qb, afW, afb, axW, axb,
        mW1, mb1, mW2, mb2, xW, xb, featT, outp + (size_t)BB * MM * 3);
}